// SimpleGNN_28578712387811
// MI455X (gfx1250) — compile-verified
//
#include <hip/hip_runtime.h>
#include <math.h>

typedef __attribute__((ext_vector_type(2))) float v2f;
typedef __attribute__((ext_vector_type(8))) float v8f;

#ifndef DIV_UP
#define DIV_UP(a, b) (((a) + (b) - 1) / (b))
#endif

// ---------------- degree / normalization ----------------

__global__ void gcn_init_deg(float* __restrict__ deg, int N) {
    int i = blockIdx.x * blockDim.x + threadIdx.x;
    if (i < N) deg[i] = 1.0f;  // self-loop contributes 1
}

__global__ void gcn_count_deg(const int* __restrict__ dst, float* __restrict__ deg, int E) {
    int e = blockIdx.x * blockDim.x + threadIdx.x;
    if (e < E) atomicAdd(&deg[dst[e]], 1.0f);
}

__global__ void gcn_dinv(float* __restrict__ deg, int N) {
    int i = blockIdx.x * blockDim.x + threadIdx.x;
    if (i < N) deg[i] = rsqrtf(deg[i]);  // deg >= 1 always (self-loops)
}

// ---------------- layer-1 GEMM: h = x @ W1 via fp32 WMMA ----------------
// One wave per 16x16 output tile. K=16 split into 4 steps of
// V_WMMA_F32_16X16X4_F32 (native fp32 -> bit-compatible precision class with
// the fp32 reference). A layout (ISA 7.12.2, 32-bit A 16x4): lane = M,
// VGPR pair holds K={0,1} for lanes 0-15, K={2,3} for lanes 16-31.
// B layout mirrored: lane = N. C/D: VGPR v -> row M = v + 8*(lane>=16), N = lane%16.
__global__ void gcn_gemm16_wmma(const float* __restrict__ x,
                                const float* __restrict__ W,
                                float* __restrict__ h, int tiles) {
    int wave = (blockIdx.x * blockDim.x + threadIdx.x) >> 5;
    if (wave >= tiles) return;  // uniform across the wave: EXEC stays all-1s
    int lane = threadIdx.x & 31;
    int half = lane >> 4;       // 0: lanes 0-15, 1: lanes 16-31
    int n    = lane & 15;       // M index for A loads, N index for B/C
    int row0 = wave * 16;

    const float* arow = x + (row0 + n) * 16 + 2 * half;
    v8f c = {};
#pragma unroll
    for (int kk = 0; kk < 4; ++kk) {
        v2f a = *(const v2f*)(arow + 4 * kk);           // global_load_b64
        v2f b;
        b.x = W[(4 * kk + 2 * half + 0) * 16 + n];
        b.y = W[(4 * kk + 2 * half + 1) * 16 + n];
        c = __builtin_amdgcn_wmma_f32_16x16x4_f32(false, a, false, b,
                                                  (short)0, c, false, false);
    }
#pragma unroll
    for (int v = 0; v < 8; ++v) {
        h[(row0 + v + 8 * half) * 16 + n] = c[v];       // 16 lanes -> 64B coalesced
    }
}

// scalar fallback for a ragged tail (N % 16 != 0); unused for N = 100000
__global__ void gcn_gemm16_tail(const float* __restrict__ x,
                                const float* __restrict__ W,
                                float* __restrict__ h, int rowStart, int N) {
    int t = blockIdx.x * blockDim.x + threadIdx.x;
    int r = rowStart + (t >> 4);
    int cidx = t & 15;
    if (r >= N) return;
    float acc = 0.0f;
#pragma unroll
    for (int k = 0; k < 16; ++k) acc += x[r * 16 + k] * W[k * 16 + cidx];
    h[r * 16 + cidx] = acc;
}

// ---------------- layer-1 aggregation ----------------

// out1[i][:] = h[i][:] * dinv[i]^2   (self-loop message; also initializes out1)
__global__ void gcn_self1(const float* __restrict__ h, const float* __restrict__ dinv,
                          float* __restrict__ out1, int N) {
    int i = blockIdx.x * blockDim.x + threadIdx.x;
    if (i >= N) return;
    float s = dinv[i] * dinv[i];
    const float4* hv = (const float4*)(h + i * 16);
    float4* ov = (float4*)(out1 + i * 16);
#pragma unroll
    for (int g = 0; g < 4; ++g) {
        float4 v = hv[g];
        ov[g] = make_float4(v.x * s, v.y * s, v.z * s, v.w * s);
    }
}

// 4 threads per edge, one float4 feature group each -> 4 f32 scatter-atomics.
// h (6.4 MB) is L2-resident on MI455X (192 MB L2): gather + atomics stay in L2.
__global__ void gcn_agg1(const int* __restrict__ src, const int* __restrict__ dst,
                         const float* __restrict__ h, const float* __restrict__ dinv,
                         float* __restrict__ out1, int E) {
    int t = blockIdx.x * blockDim.x + threadIdx.x;
    int e = t >> 2;
    if (e >= E) return;
    int g = t & 3;
    int s = src[e], d = dst[e];
    float norm = dinv[s] * dinv[d];
    float4 v = ((const float4*)(h + s * 16))[g];
    float* o = out1 + d * 16 + g * 4;
    atomicAdd(o + 0, v.x * norm);
    atomicAdd(o + 1, v.y * norm);
    atomicAdd(o + 2, v.z * norm);
    atomicAdd(o + 3, v.w * norm);
}

// h2[i] = relu(out1[i] + b1) . W2   (fused bias + relu + layer-2 linear)
__global__ void gcn_finish1(const float* __restrict__ out1, const float* __restrict__ b1,
                            const float* __restrict__ W2, float* __restrict__ h2, int N) {
    int i = blockIdx.x * blockDim.x + threadIdx.x;
    if (i >= N) return;
    float acc = 0.0f;
#pragma unroll
    for (int f = 0; f < 16; ++f) {
        float z = out1[i * 16 + f] + b1[f];
        z = fmaxf(z, 0.0f);
        acc += z * W2[f];
    }
    h2[i] = acc;
}

// ---------------- layer-2 aggregation ----------------

// out[i] = h2[i] * dinv[i]^2 + b2    (self-loop + bias; initializes out)
__global__ void gcn_self2(const float* __restrict__ h2, const float* __restrict__ dinv,
                          const float* __restrict__ b2, float* __restrict__ out, int N) {
    int i = blockIdx.x * blockDim.x + threadIdx.x;
    if (i >= N) return;
    out[i] = h2[i] * dinv[i] * dinv[i] + b2[0];
}

__global__ void gcn_agg2(const int* __restrict__ src, const int* __restrict__ dst,
                         const float* __restrict__ h2, const float* __restrict__ dinv,
                         float* __restrict__ out, int E) {
    int e = blockIdx.x * blockDim.x + threadIdx.x;
    if (e >= E) return;
    int s = src[e], d = dst[e];
    atomicAdd(&out[d], h2[s] * dinv[s] * dinv[d]);
}

// ---------------- driver ----------------

extern "C" void kernel_launch(void* const* d_in, const int* in_sizes, int n_in,
                              void* d_out, int out_size, void* d_ws, size_t ws_size,
                              hipStream_t stream) {
    const float* x  = (const float*)d_in[0];   // [N,16]
    const int*   ei = (const int*)d_in[1];     // [2,E]
    const float* W1 = (const float*)d_in[2];   // [16,16]
    const float* b1 = (const float*)d_in[3];   // [16]
    const float* W2 = (const float*)d_in[4];   // [16,1]
    const float* b2 = (const float*)d_in[5];   // [1]

    const int N = in_sizes[0] / 16;
    const int E = in_sizes[1] / 2;
    const int* src = ei;
    const int* dst = ei + E;

    // workspace layout (floats): dinv[N] | h[16N] | out1[16N] | h2[N]  = 34N
    float* ws   = (float*)d_ws;
    float* dinv = ws;
    float* h    = ws + N;
    float* out1 = ws + N + 16 * (size_t)N;
    float* h2   = ws + N + 32 * (size_t)N;
    float* out  = (float*)d_out;

    const int B = 256;

    // degrees -> dinv (in place)
    gcn_init_deg<<<DIV_UP(N, B), B, 0, stream>>>(dinv, N);
    gcn_count_deg<<<DIV_UP(E, B), B, 0, stream>>>(dst, dinv, E);
    gcn_dinv<<<DIV_UP(N, B), B, 0, stream>>>(dinv, N);

    // layer-1 GEMM on the fp32 WMMA pipe: one wave per 16x16 tile
    int tiles = N / 16;
    if (tiles > 0) {
        int wavesPerBlock = 4;  // 128 threads
        gcn_gemm16_wmma<<<DIV_UP(tiles, wavesPerBlock), wavesPerBlock * 32, 0, stream>>>(
            x, W1, h, tiles);
    }
    int rem = N - tiles * 16;
    if (rem > 0) {
        gcn_gemm16_tail<<<DIV_UP(rem * 16, B), B, 0, stream>>>(x, W1, h, tiles * 16, N);
    }

    // layer-1 aggregate: self-loop init then edge scatter-atomics (L2-resident)
    gcn_self1<<<DIV_UP(N, B), B, 0, stream>>>(h, dinv, out1, N);
    gcn_agg1<<<DIV_UP(E * 4, B), B, 0, stream>>>(src, dst, h, dinv, out1, E);

    // fused bias+relu+layer-2 linear
    gcn_finish1<<<DIV_UP(N, B), B, 0, stream>>>(out1, b1, W2, h2, N);

    // layer-2 aggregate
    gcn_self2<<<DIV_UP(N, B), B, 0, stream>>>(h2, dinv, b2, out, N);
    gcn_agg2<<<DIV_UP(E, B), B, 0, stream>>>(src, dst, h2, dinv, out, E);
}